// GNNEncoder_27118423507680
// MI455X (gfx1250) — compile-verified
//
#include <hip/hip_runtime.h>

typedef __attribute__((ext_vector_type(2))) float v2f;
typedef __attribute__((ext_vector_type(8))) float v8f;

#define DIM 128
#define LDSP 132   // padded row stride (floats): kills LDS bank conflicts, keeps 16B align (132*4=528=33*16)

// ---------------- graph preprocessing ----------------

__global__ __launch_bounds__(256) void k_deg_init(float* __restrict__ deg, int n) {
    int i = blockIdx.x * 256 + threadIdx.x;
    if (i < n) deg[i] = 1.0f;            // self-loop weight
}

__global__ __launch_bounds__(256) void k_deg_edges(const int* __restrict__ dst,
                                                   const float* __restrict__ ew,
                                                   float* __restrict__ deg, int e) {
    int i = blockIdx.x * 256 + threadIdx.x;
    if (i < e) unsafeAtomicAdd(&deg[dst[i]], ew[i]);   // global_atomic_add_f32
}

__global__ __launch_bounds__(256) void k_dinv(const float* __restrict__ deg,
                                              float* __restrict__ dinv, int n) {
    int i = blockIdx.x * 256 + threadIdx.x;
    if (i < n) {
        float d = deg[i];
        dinv[i] = (d > 0.0f) ? rsqrtf(d) : 0.0f;
    }
}

__global__ __launch_bounds__(256) void k_norm(const int* __restrict__ src,
                                              const int* __restrict__ dst,
                                              const float* __restrict__ ew,
                                              const float* __restrict__ dinv,
                                              float* __restrict__ nrm, int e) {
    int i = blockIdx.x * 256 + threadIdx.x;
    if (i < e) nrm[i] = dinv[src[i]] * ew[i] * dinv[dst[i]];
}

// ---------------- fp32 WMMA GEMM: h = x @ W ; accum = h * dinv^2 ----------------
// Block = 256 threads = 8 waves. Block owns a 16-row stripe of x; wave w owns
// column tile [16w, 16w+16). K=128 -> 32 chained v_wmma_f32_16x16x4_f32.
// Epilogue repacks C tiles through LDS for coalesced b128 writeback.

__global__ __launch_bounds__(256) void k_gemm(const float* __restrict__ x,
                                              const float* __restrict__ W,
                                              const float* __restrict__ dinv,
                                              float* __restrict__ h,
                                              float* __restrict__ accum, int n) {
    __shared__ float Ws[DIM * LDSP];   // W transposed: Ws[col*LDSP + k]
    __shared__ float Xs[16 * LDSP];    // x tile (A), later reused as C tile
    __shared__ float Ds[16];           // dinv for the 16 rows of this stripe

    const int tid     = threadIdx.x;
    const int rowBase = blockIdx.x * 16;

    // stage W transposed into LDS: b128 global reads, scattered LDS writes
    for (int i = tid; i < DIM * 32; i += 256) {      // i indexes float4 of W
        int k  = i >> 5;                             // W row (K index)
        int c4 = i & 31;                             // float4 within row
        float4 v = *(const float4*)&W[(size_t)k * DIM + c4 * 4];
        int cb = c4 * 4;
        Ws[(cb + 0) * LDSP + k] = v.x;
        Ws[(cb + 1) * LDSP + k] = v.y;
        Ws[(cb + 2) * LDSP + k] = v.z;
        Ws[(cb + 3) * LDSP + k] = v.w;
    }
    // stage 16x128 x tile: b128 reads/writes (clamp rows for safety)
    for (int i = tid; i < 16 * 32; i += 256) {
        int r  = i >> 5;
        int c4 = i & 31;
        int rr = rowBase + r; if (rr >= n) rr = n - 1;
        float4 v = *(const float4*)&x[(size_t)rr * DIM + c4 * 4];
        *(float4*)&Xs[r * LDSP + c4 * 4] = v;
    }
    // stage dinv for this stripe
    if (tid < 16) {
        int rr = rowBase + tid; if (rr >= n) rr = n - 1;
        Ds[tid] = dinv[rr];
    }
    __syncthreads();

    const int wave    = tid >> 5;
    const int lane    = tid & 31;
    const int colBase = wave * 16;
    const int l16     = lane & 15;
    const int hi      = lane >> 4;     // 0: K pair {k,k+1}, 1: K pair {k+2,k+3}

    v8f c = {};
    #pragma unroll 4
    for (int k = 0; k < DIM; k += 4) {
        const int ka = k + hi * 2;
        v2f a, b;
        const float2 av = *(const float2*)&Xs[l16 * LDSP + ka];                 // ds_load_b64
        const float2 bv = *(const float2*)&Ws[(colBase + l16) * LDSP + ka];     // ds_load_b64
        a.x = av.x; a.y = av.y;
        b.x = bv.x; b.y = bv.y;
        c = __builtin_amdgcn_wmma_f32_16x16x4_f32(false, a, false, b,
                                                  (short)0, c, false, false);
    }

    // repack: scatter each wave's 16x16 C tile into LDS (Xs is dead as A now)
    __syncthreads();
    const int col = colBase + l16;
    #pragma unroll
    for (int j = 0; j < 8; ++j) {
        // C/D layout: VGPR j -> tile row (j + 8*hi), lane&15 -> col
        Xs[(j + hi * 8) * LDSP + col] = c[j];
    }
    __syncthreads();

    // coalesced b128 writeback: 512 float4s, 2 per thread
    for (int i = tid; i < 16 * 32; i += 256) {
        int r   = i >> 5;
        int c4  = i & 31;
        int row = rowBase + r;
        if (row < n) {
            float4 v = *(const float4*)&Xs[r * LDSP + c4 * 4];
            float di = Ds[r];
            float s  = di * di;
            size_t idx = (size_t)row * DIM + c4 * 4;
            *(float4*)&h[idx] = v;
            float4 a; a.x = v.x * s; a.y = v.y * s; a.z = v.z * s; a.w = v.w * s;
            *(float4*)&accum[idx] = a;   // seed with self-loop term
        }
    }
}

// ---------------- edge scatter: accum[dst] += norm * h[src] ----------------
// One edge per wave32: lane l handles features 4l..4l+3 (float4 = full 512B row/wave).

__global__ __launch_bounds__(256) void k_scatter(const int* __restrict__ src,
                                                 const int* __restrict__ dst,
                                                 const float* __restrict__ nrm,
                                                 const float* __restrict__ h,
                                                 float* __restrict__ accum, int e) {
    int wid  = (blockIdx.x * 256 + threadIdx.x) >> 5;
    int lane = threadIdx.x & 31;
    if (wid < e) {
        int s = src[wid], d = dst[wid];
        float w = nrm[wid];
        const float4 hv = *(const float4*)&h[(size_t)s * DIM + lane * 4];
        float* out = &accum[(size_t)d * DIM + lane * 4];
        unsafeAtomicAdd(out + 0, hv.x * w);
        unsafeAtomicAdd(out + 1, hv.y * w);
        unsafeAtomicAdd(out + 2, hv.z * w);
        unsafeAtomicAdd(out + 3, hv.w * w);
    }
}

// ---------------- bias + ReLU (in-place capable) ----------------

__global__ __launch_bounds__(256) void k_finalize(const float* __restrict__ accum,
                                                  const float* __restrict__ bias,
                                                  float* __restrict__ out, size_t total) {
    size_t i = (size_t)blockIdx.x * 256 + threadIdx.x;
    if (i < total) {
        float v = accum[i] + bias[i & (DIM - 1)];
        out[i] = v > 0.0f ? v : 0.0f;
    }
}

// ---------------- launcher ----------------

extern "C" void kernel_launch(void* const* d_in, const int* in_sizes, int n_in,
                              void* d_out, int out_size, void* d_ws, size_t ws_size,
                              hipStream_t stream) {
    const float* x  = (const float*)d_in[0];
    const int*   ei = (const int*)d_in[1];     // [2, E]
    const float* ew = (const float*)d_in[2];
    const float* W1 = (const float*)d_in[3];
    const float* b1 = (const float*)d_in[4];
    const float* W2 = (const float*)d_in[5];
    const float* b2 = (const float*)d_in[6];
    const float* W3 = (const float*)d_in[7];
    const float* b3 = (const float*)d_in[8];

    const int n = in_sizes[0] / DIM;
    const int e = in_sizes[2];
    const int* src = ei;
    const int* dst = ei + e;

    // workspace layout (floats): deg[n] dinv[n] norm[e] h[n*D] accA[n*D] accB[n*D]
    float* deg  = (float*)d_ws;
    float* dinv = deg  + n;
    float* nrm  = dinv + n;
    float* h    = nrm  + e;
    float* accA = h    + (size_t)n * DIM;
    float* accB = accA + (size_t)n * DIM;

    const int    nb       = (n + 255) / 256;
    const int    ebk      = (e + 255) / 256;
    const int    rowTiles = (n + 15) / 16;
    const int    scatB    = (int)(((size_t)e * 32 + 255) / 256);
    const size_t total    = (size_t)n * DIM;
    const int    finB     = (int)((total + 255) / 256);

    // graph normalization (layer-invariant)
    k_deg_init <<<nb,   256, 0, stream>>>(deg, n);
    k_deg_edges<<<ebk,  256, 0, stream>>>(dst, ew, deg, e);
    k_dinv     <<<nb,   256, 0, stream>>>(deg, dinv, n);
    k_norm     <<<ebk,  256, 0, stream>>>(src, dst, ew, dinv, nrm, e);

    // layer 1: x -> accA
    k_gemm     <<<rowTiles, 256, 0, stream>>>(x, W1, dinv, h, accA, n);
    k_scatter  <<<scatB,    256, 0, stream>>>(src, dst, nrm, h, accA, e);
    k_finalize <<<finB,     256, 0, stream>>>(accA, b1, accA, total);

    // layer 2: accA -> accB
    k_gemm     <<<rowTiles, 256, 0, stream>>>(accA, W2, dinv, h, accB, n);
    k_scatter  <<<scatB,    256, 0, stream>>>(src, dst, nrm, h, accB, e);
    k_finalize <<<finB,     256, 0, stream>>>(accB, b2, accB, total);

    // layer 3: accB -> d_out
    k_gemm     <<<rowTiles, 256, 0, stream>>>(accB, W3, dinv, h, accA, n);
    k_scatter  <<<scatB,    256, 0, stream>>>(src, dst, nrm, h, accA, e);
    k_finalize <<<finB,     256, 0, stream>>>(accA, b3, (float*)d_out, total);
}